// GALSTM_65644280152384
// MI455X (gfx1250) — compile-verified
//
#include <hip/hip_runtime.h>
#include <math.h>

typedef __attribute__((ext_vector_type(2))) float v2f;
typedef __attribute__((ext_vector_type(8))) float v8f;

#define BB 4
#define SSS 12
#define NNN 307
#define RRT (BB*SSS*NNN)   // 14736 rows (b,s,n)
#define LLT (BB*NNN)       // 1228 rows (b,n)
#define PPP 12

static __device__ __forceinline__ int imin(int a, int b) { return a < b ? a : b; }
static __device__ __forceinline__ float sigf(float x) { return 1.0f / (1.0f + __expf(-x)); }

// ---------------- WMMA f32 16x16x4 wrapper ----------------
static __device__ __forceinline__ v8f wmma4(v2f a, v2f b, v8f c) {
  return __builtin_amdgcn_wmma_f32_16x16x4_f32(false, a, false, b, 0, c, false, false);
}

// One 16x16 C tile per wave: branch-free inner loop.
//  - M/N edges: clamp the per-lane base row/col once; garbage lands only in
//    C rows/cols that the guarded store never writes.
//  - K edge: peeled single tail step using clamped addresses + value selects
//    (v_cndmask), never exec-mask branches. EXEC stays all-ones for WMMA.
static __device__ __forceinline__ v8f gemm_tile(const float* A, int lda, int Mrem,
                                                const float* B, int ldb, int Nrem,
                                                int Kdim, int transB, v8f acc) {
  const int lane = threadIdx.x & 31;
  const int half = lane >> 4;
  const int q = lane & 15;
  const int qa = (q < Mrem) ? q : 0;
  const int qb = (q < Nrem) ? q : 0;
  const float* Ar = A + (long)qa * lda;
  const int Kmain = Kdim & ~3;
  if (transB) {
    const float* Br = B + (long)qb * ldb;
    for (int k = 0; k < Kmain; k += 4) {
      const int k0 = k + 2 * half;
      v2f a, b;
      a.x = Ar[k0]; a.y = Ar[k0 + 1];
      b.x = Br[k0]; b.y = Br[k0 + 1];
      acc = wmma4(a, b, acc);
    }
    if (Kmain < Kdim) {
      const int k0 = Kmain + 2 * half, k1 = k0 + 1;
      const int k0c = imin(k0, Kdim - 1), k1c = imin(k1, Kdim - 1);
      const float a0 = Ar[k0c], a1 = Ar[k1c];
      const float b0 = Br[k0c], b1 = Br[k1c];
      v2f a, b;
      a.x = (k0 < Kdim) ? a0 : 0.0f;
      a.y = (k1 < Kdim) ? a1 : 0.0f;
      b.x = (k0 < Kdim) ? b0 : 0.0f;
      b.y = (k1 < Kdim) ? b1 : 0.0f;
      acc = wmma4(a, b, acc);
    }
  } else {
    for (int k = 0; k < Kmain; k += 4) {
      const int k0 = k + 2 * half;
      v2f a, b;
      a.x = Ar[k0]; a.y = Ar[k0 + 1];
      b.x = B[(long)k0 * ldb + qb];
      b.y = B[(long)(k0 + 1) * ldb + qb];
      acc = wmma4(a, b, acc);
    }
    if (Kmain < Kdim) {
      const int k0 = Kmain + 2 * half, k1 = k0 + 1;
      const int k0c = imin(k0, Kdim - 1), k1c = imin(k1, Kdim - 1);
      const float a0 = Ar[k0c], a1 = Ar[k1c];
      const float b0 = B[(long)k0c * ldb + qb], b1 = B[(long)k1c * ldb + qb];
      v2f a, b;
      a.x = (k0 < Kdim) ? a0 : 0.0f;
      a.y = (k1 < Kdim) ? a1 : 0.0f;
      b.x = (k0 < Kdim) ? b0 : 0.0f;
      b.y = (k1 < Kdim) ? b1 : 0.0f;
      acc = wmma4(a, b, acc);
    }
  }
  return acc;
}

// ---------------- generic GEMM: C = act(A@B(^T) + bias), batched ----------------
__global__ void __launch_bounds__(128)
gemm_kernel(const float* __restrict__ A, const float* __restrict__ B,
            const float* __restrict__ bias, float* __restrict__ C,
            int M, int N, int K, int lda, int ldb, int ldc,
            long sA, long sB, long sC, int transB, int act) {
  const int wave = threadIdx.x >> 5;
  const int m0 = (blockIdx.y * 4 + wave) * 16;
  if (m0 >= M) return;                       // wave-uniform
  const int n0 = blockIdx.x * 16;
  const int z = blockIdx.z;
  const int Mrem = imin(16, M - m0);
  const int Nrem = imin(16, N - n0);
  const float* Ap = A + (long)z * sA + (long)m0 * lda;
  const float* Bp = B + (long)z * sB + (transB ? (long)n0 * ldb : (long)n0);
  float* Cp = C + (long)z * sC + (long)m0 * ldc + n0;
  v8f acc = {};
  acc = gemm_tile(Ap, lda, Mrem, Bp, ldb, Nrem, K, transB, acc);
  const int lane = threadIdx.x & 31, half = lane >> 4, q = lane & 15;
#pragma unroll
  for (int r = 0; r < 8; ++r) {
    const int row = r + 8 * half;
    if (row < Mrem && q < Nrem) {
      float v = acc[r];
      if (bias) v += bias[n0 + q];
      if (act) v = fmaxf(v, 0.0f);
      Cp[(long)row * ldc + q] = v;
    }
  }
}

// ---------------- embeddings ----------------
__global__ void embed_kernel(const float* __restrict__ x, const float* __restrict__ Wx,
                             const float* __restrict__ bx, const float* __restrict__ Wos,
                             const float* __restrict__ bos, float* __restrict__ ex,
                             float* __restrict__ eos) {
  const long idx = (long)blockIdx.x * blockDim.x + threadIdx.x;
  const long total = (long)RRT * 64;
  if (idx >= total) return;
  const int e = (int)(idx & 63);
  const long r = idx >> 6;
  const float x0 = x[r * 3 + 0], x1 = x[r * 3 + 1], x2 = x[r * 3 + 2];
  ex[idx] = x0 * Wx[e] + bx[e];
  eos[idx] = x1 * Wos[e] + x2 * Wos[64 + e] + bos[e];
}

// ---------------- cheb conv: gcn(b,s) = relu(sum_k cheb_k @ tmp_k(b,s)) ----------
// All 4 waves of the block share the (307x16) B panel of tmp_k -> stage it into
// LDS with CDNA5 async global->LDS copies (ASYNCcnt), padded to 320 rows so the
// WMMA K-tail can read LDS without guards.
__global__ void __launch_bounds__(128)
cheb_kernel(const float* __restrict__ cheb, const float* __restrict__ tmp,
            float* __restrict__ gcn) {
  __shared__ __align__(16) float sB[1280 * 4];   // 320 rows x 16 cols
  const int wave = threadIdx.x >> 5;
  const int m0 = (blockIdx.y * 4 + wave) * 16;   // grid sized exactly: m0 <= 304
  const int n0 = blockIdx.x * 16;
  const int bs = blockIdx.z;                     // 0..47
  const int Mrem = imin(16, NNN - m0);
  const int lane = threadIdx.x & 31, half = lane >> 4, q = lane & 15;
  const unsigned ldsBase = (unsigned)(unsigned long long)(void*)&sB[0];
  v8f acc = {};
  for (int k = 0; k < 3; ++k) {
    __syncthreads();                             // WAR protection on sB
    const float* Bp = tmp + ((long)k * RRT + (long)bs * NNN) * 64 + n0;
    for (int c = threadIdx.x; c < 1280; c += 128) {
      const int cc = imin(c, 1227);              // clamp to last valid 16B chunk
      const int row = cc >> 2, col4 = (cc & 3) << 2;
      const unsigned long long g =
          (unsigned long long)(const void*)(Bp + (long)row * 64 + col4);
      const unsigned lo = ldsBase + (unsigned)c * 16u;
      asm volatile("global_load_async_to_lds_b128 %0, %1, off"
                   :: "v"(lo), "v"(g) : "memory");
    }
    asm volatile("s_wait_asynccnt 0x0" ::: "memory");
    __syncthreads();
    const float* Ap = cheb + ((long)k * NNN + m0) * NNN;
    const float* Ar = Ap + (long)((q < Mrem) ? q : 0) * NNN;
    for (int kk = 0; kk < 304; kk += 4) {        // branch-free main loop
      const int k0 = kk + 2 * half;
      v2f a, b;
      a.x = Ar[k0]; a.y = Ar[k0 + 1];
      b.x = sB[k0 * 16 + q]; b.y = sB[(k0 + 1) * 16 + q];
      acc = wmma4(a, b, acc);
    }
    {                                            // K tail 304..306
      const int k0 = 304 + 2 * half, k1 = k0 + 1;
      const float a0 = Ar[imin(k0, 306)];
      const float a1 = Ar[imin(k1, 306)];
      v2f a, b;
      a.x = (k0 < NNN) ? a0 : 0.0f;
      a.y = (k1 < NNN) ? a1 : 0.0f;
      b.x = sB[k0 * 16 + q];                     // rows 307..319 = pad, a masks them
      b.y = sB[k1 * 16 + q];
      acc = wmma4(a, b, acc);
    }
  }
  float* Cp = gcn + ((long)bs * NNN + m0) * 64 + n0;
#pragma unroll
  for (int r = 0; r < 8; ++r) {
    const int row = r + 8 * half;
    if (row < Mrem) Cp[(long)row * 64 + q] = fmaxf(acc[r], 0.0f);
  }
}

// ---------------- misc elementwise ----------------
__global__ void zero_kernel(float* __restrict__ p, long n) {
  const long i = (long)blockIdx.x * blockDim.x + threadIdx.x;
  if (i < n) p[i] = 0.0f;
}

// ---------------- LSTM step (one t): gates via WMMA h@V in LDS ----------------
__global__ void __launch_bounds__(128)
lstm_step_kernel(const float* __restrict__ gx,  // (B,S,N,256)
                 const float* __restrict__ gg,  // (B,S,N,128)
                 const float* __restrict__ Vx,  // (64,256)
                 const float* __restrict__ Vg,  // (64,128)
                 float* __restrict__ h, float* __restrict__ c,
                 float* __restrict__ hs,        // (B,S,N,64)
                 int t) {
  __shared__ float sh[16 * 64];
  __shared__ float sp[16 * 384];
  const int r0 = blockIdx.x * 16;
  for (int i = threadIdx.x; i < 16 * 64; i += blockDim.x) {
    const int rr = i >> 6, cc = i & 63;
    sh[i] = (r0 + rr < LLT) ? h[(long)(r0 + rr) * 64 + cc] : 0.0f;
  }
  __syncthreads();
  const int wave = threadIdx.x >> 5;
  const int lane = threadIdx.x & 31, half = lane >> 4, q = lane & 15;
  for (int nt = wave; nt < 24; nt += 4) {  // 16 Vx tiles + 8 Vg tiles
    const float* Bp;
    int ldb, n0;
    if (nt < 16) { Bp = Vx + nt * 16; ldb = 256; n0 = nt * 16; }
    else         { Bp = Vg + (nt - 16) * 16; ldb = 128; n0 = 256 + (nt - 16) * 16; }
    v8f acc = {};
    acc = gemm_tile(sh, 64, 16, Bp, ldb, 16, 64, 0, acc);  // full tile: no guards
#pragma unroll
    for (int r = 0; r < 8; ++r) sp[(r + 8 * half) * 384 + n0 + q] = acc[r];
  }
  __syncthreads();
  for (int i = threadIdx.x; i < 16 * 64; i += blockDim.x) {
    const int rr = i >> 6, hh = i & 63;
    const int rg = r0 + rr;
    if (rg >= LLT) continue;
    const int b = rg / NNN, n = rg % NNN;
    const long gxb = (((long)b * SSS + t) * NNN + n) * 256;
    const long ggb = (((long)b * SSS + t) * NNN + n) * 128;
    const float ft = sigf(gx[gxb + hh]       + sp[rr * 384 + hh]);
    const float it = sigf(gx[gxb + 64 + hh]  + sp[rr * 384 + 64 + hh]);
    const float ot = sigf(gx[gxb + 128 + hh] + sp[rr * 384 + 128 + hh]);
    const float ut = sigf(gx[gxb + 192 + hh] + sp[rr * 384 + 192 + hh]);
    const float gf = sigf(gg[ggb + hh]       + sp[rr * 384 + 256 + hh]);
    const float gu = sigf(gg[ggb + 64 + hh]  + sp[rr * 384 + 320 + hh]);
    const long ci = (long)rg * 64 + hh;
    const float cv = gf * ft * c[ci] + gu * it * ut;
    const float hv = ot * tanhf(cv);
    c[ci] = cv;
    h[ci] = hv;
    hs[(((long)b * SSS + t) * NNN + n) * 64 + hh] = hv;
  }
}

// ---------------- time conv (1,3), pad 1 over s ----------------
__global__ void tconv_kernel(const float* __restrict__ hid, const float* __restrict__ Wtc,
                             const float* __restrict__ btc, float* __restrict__ tc) {
  const long idx = (long)blockIdx.x * blockDim.x + threadIdx.x;
  const long total = (long)RRT * 64;
  if (idx >= total) return;
  const int f = (int)(idx & 63);
  long r = idx >> 6;
  const int n = (int)(r % NNN); r /= NNN;
  const int s = (int)(r % SSS);
  const int b = (int)(r / SSS);
  float sum = btc[f];
  for (int dt = 0; dt < 3; ++dt) {
    const int ss = s + dt - 1;
    if (ss < 0 || ss >= SSS) continue;
    const float* hp = hid + (((long)b * SSS + ss) * NNN + n) * 64;
    const float* wp = Wtc + (long)f * 64 * 3 + dt;  // W_tc[f][hh][0][dt]
    float acc = 0.0f;
    for (int hh = 0; hh < 64; ++hh) acc += hp[hh] * wp[hh * 3];
    sum += acc;
  }
  tc[idx] = sum;
}

// ---------------- attention packing ----------------
__global__ void pack_q_kernel(const float* __restrict__ tc, float* __restrict__ qsrc) {
  const long idx = (long)blockIdx.x * blockDim.x + threadIdx.x;
  const long total = (long)LLT * 64;
  if (idx >= total) return;
  const int cc = (int)(idx & 63);
  const long r = idx >> 6;
  const int n = (int)(r % NNN), b = (int)(r / NNN);
  qsrc[idx] = tc[(((long)b * SSS + (SSS - 1)) * NNN + n) * 64 + cc];  // last time step
}

__global__ void pack_kv_kernel(const float* __restrict__ ex, float* __restrict__ kv) {
  const long idx = (long)blockIdx.x * blockDim.x + threadIdx.x;
  const long total = (long)SSS * LLT * 64;
  if (idx >= total) return;
  const int cc = (int)(idx & 63);
  long r = idx >> 6;
  const int n = (int)(r % NNN); r /= NNN;
  const int b = (int)(r % BB);
  const int s = (int)(r / BB);
  kv[idx] = ex[(((long)b * SSS + s) * NNN + n) * 64 + cc];  // (s, b, n, c)
}

// ---------------- flash attention: 48 instances (s,head), hd=16 --------------
// L=1228 -> key chunks are 16,...,16,12: every K-dim is a multiple of 4, so the
// WMMA loops never take the tail path here.
__global__ void __launch_bounds__(128)
attn_kernel(const float* __restrict__ Qh,   // (L,64)
            const float* __restrict__ Kh,   // (S,L,64)
            const float* __restrict__ Vh,   // (S,L,64)
            float* __restrict__ ctx) {      // (L,S,64)
  __shared__ float sP[4][256];
  const int wave = threadIdx.x >> 5;
  const int lane = threadIdx.x & 31, half = lane >> 4, q = lane & 15;
  const int inst = blockIdx.y;               // s*4 + head
  const int s = inst >> 2, head = inst & 3;
  const int l0 = (blockIdx.x * 4 + wave) * 16;
  const int Mrem = imin(16, LLT - l0);       // may be <=0 for trailing waves
  const int l0c = (Mrem > 0) ? l0 : 0;       // keep base pointer in-bounds
  const float* Qb = Qh + (long)l0c * 64 + head * 16;
  const float* Kb = Kh + (long)s * LLT * 64 + head * 16;
  const float* Vb = Vh + (long)s * LLT * 64 + head * 16;
  v8f o = {};
  float mrow[8], lrow[8];
#pragma unroll
  for (int r = 0; r < 8; ++r) { mrow[r] = -1e30f; lrow[r] = 0.0f; }
  const float scale = 0.25f;                 // 1/sqrt(16)
  for (int m0 = 0; m0 < LLT; m0 += 16) {
    const int Krem = imin(16, LLT - m0);
    v8f sc = {};
    sc = gemm_tile(Qb, 64, Mrem, Kb + (long)m0 * 64, 64, Krem, 16, 1, sc);
#pragma unroll
    for (int r = 0; r < 8; ++r) {
      float sv = (q < Krem) ? sc[r] * scale : -1e30f;
      float mx = sv;
      for (int off = 1; off < 16; off <<= 1) mx = fmaxf(mx, __shfl_xor(mx, off));
      const float mnew = fmaxf(mrow[r], mx);
      const float p = __expf(sv - mnew);
      float psum = p;
      for (int off = 1; off < 16; off <<= 1) psum += __shfl_xor(psum, off);
      const float alpha = __expf(mrow[r] - mnew);
      lrow[r] = lrow[r] * alpha + psum;
      o[r] *= alpha;
      mrow[r] = mnew;
      sc[r] = p;
    }
#pragma unroll
    for (int r = 0; r < 8; ++r) sP[wave][(r + 8 * half) * 16 + q] = sc[r];
    __syncthreads();
    o = gemm_tile(&sP[wave][0], 16, 16, Vb + (long)m0 * 64, 64, 16, Krem, 0, o);
    __syncthreads();
  }
#pragma unroll
  for (int r = 0; r < 8; ++r) {
    const int row = r + 8 * half;
    if (row < Mrem) ctx[(((long)(l0 + row)) * SSS + s) * 64 + head * 16 + q] = o[r] / lrow[r];
  }
}

// ---------------- final: out[b,p,n] = sum_{s,f} (res+ffa)*W_fin + b_fin -------
__global__ void final_kernel(const float* __restrict__ resb,  // (B,S,N,64)
                             const float* __restrict__ ffab,  // (B*N,S,64)
                             const float* __restrict__ Wfin,  // (P,S,1,64)
                             const float* __restrict__ bfin, float* __restrict__ out) {
  const long idx = (long)blockIdx.x * blockDim.x + threadIdx.x;
  const long total = (long)BB * PPP * NNN;
  if (idx >= total) return;
  const int n = (int)(idx % NNN);
  const long r = idx / NNN;
  const int p = (int)(r % PPP), b = (int)(r / PPP);
  float sum = bfin[p];
  for (int s = 0; s < SSS; ++s) {
    const float* rp = resb + (((long)b * SSS + s) * NNN + n) * 64;
    const float* fp = ffab + (((long)b * NNN + n) * SSS + s) * 64;
    const float* wp = Wfin + ((long)p * SSS + s) * 64;
    float acc = 0.0f;
    for (int f = 0; f < 64; ++f) acc += (rp[f] + fp[f]) * wp[f];
    sum += acc;
  }
  out[idx] = sum;
}

// ---------------- host side ----------------
static inline void launch_gemm(hipStream_t st, const float* A, const float* B,
                               const float* bias, float* C, int M, int N, int K,
                               int lda, int ldb, int ldc, int batch, long sA, long sB,
                               long sC, int transB, int act) {
  const int mt = (M + 15) / 16;
  dim3 grid((N + 15) / 16, (mt + 3) / 4, batch);
  gemm_kernel<<<grid, 128, 0, st>>>(A, B, bias, C, M, N, K, lda, ldb, ldc, sA, sB, sC,
                                    transB, act);
}

extern "C" void kernel_launch(void* const* d_in, const int* in_sizes, int n_in,
                              void* d_out, int out_size, void* d_ws, size_t ws_size,
                              hipStream_t stream) {
  (void)in_sizes; (void)n_in; (void)out_size; (void)ws_size;
  const float* x     = (const float*)d_in[0];
  const float* cheb  = (const float*)d_in[1];
  const float* theta = (const float*)d_in[2];
  const float* Wex   = (const float*)d_in[3];
  const float* bex   = (const float*)d_in[4];
  const float* Wos   = (const float*)d_in[5];
  const float* bos   = (const float*)d_in[6];
  const float* Ux    = (const float*)d_in[7];
  const float* Vx    = (const float*)d_in[8];
  const float* bx    = (const float*)d_in[9];
  const float* Ug    = (const float*)d_in[10];
  const float* Vg    = (const float*)d_in[11];
  const float* bg    = (const float*)d_in[12];
  const float* Wff1  = (const float*)d_in[13];
  const float* bff1  = (const float*)d_in[14];
  const float* Wff2  = (const float*)d_in[15];
  const float* bff2  = (const float*)d_in[16];
  const float* Wtc   = (const float*)d_in[17];
  const float* btc   = (const float*)d_in[18];
  const float* Wrc   = (const float*)d_in[19];
  const float* brc   = (const float*)d_in[20];
  const float* Win   = (const float*)d_in[21];
  const float* bin   = (const float*)d_in[22];
  const float* Wout  = (const float*)d_in[23];
  const float* bout  = (const float*)d_in[24];
  const float* Wfa1  = (const float*)d_in[25];
  const float* bfa1  = (const float*)d_in[26];
  const float* Wfa2  = (const float*)d_in[27];
  const float* bfa2  = (const float*)d_in[28];
  const float* Wfin  = (const float*)d_in[29];
  const float* bfin  = (const float*)d_in[30];

  float* ws = (float*)d_ws;
  const long U = (long)RRT * 64;           // 943,104 floats per slot
  const long L64 = (long)LLT * 64;
  // slot lifetimes chosen so total ws = 12 slots + 4 small buffers (~47 MB)
  float* ex   = ws + 0 * U;
  float* eos  = ws + 1 * U;
  float* tmpx = ws + 2 * U;   // slots 2..4
  float* tmpo = ws + 5 * U;   // slots 5..7
  float* gcnx = ws + 8 * U;
  float* gcno = ws + 1 * U;   // reuse eos
  float* gxb  = ws + 2 * U;   // slots 2..5 (tmp dead)
  float* ggb  = ws + 6 * U;   // slots 6..7
  float* hsb  = ws + 9 * U;
  float* ff1b = ws + 10 * U;
  float* hid2 = ws + 11 * U;
  float* tcb  = ws + 8 * U;   // reuse gcnx
  float* kvb  = ws + 9 * U;   // reuse hs
  float* Khb  = ws + 0 * U;   // reuse ex
  float* Vhb  = ws + 1 * U;   // reuse gcno
  float* ctxb = ws + 2 * U;   // reuse gx
  float* dsab = ws + 3 * U;
  float* fa1b = ws + 4 * U;
  float* ffab = ws + 5 * U;
  float* resb = ws + 6 * U;   // reuse gg
  float* smallb = ws + 12 * U;
  float* hb   = smallb;
  float* cb   = smallb + L64;
  float* qsrc = smallb + 2 * L64;
  float* Qhb  = smallb + 3 * L64;

  const int T256 = 256;
  const long tot = U;  // RRT*64

  // 1) embeddings
  embed_kernel<<<dim3((unsigned)((tot + T256 - 1) / T256)), T256, 0, stream>>>(
      x, Wex, bex, Wos, bos, ex, eos);

  // 2) tmp_k = emb @ theta_k  (batched over k)
  launch_gemm(stream, ex, theta, nullptr, tmpx, RRT, 64, 64, 64, 64, 64, 3, 0, 4096, U, 0, 0);
  launch_gemm(stream, eos, theta, nullptr, tmpo, RRT, 64, 64, 64, 64, 64, 3, 0, 4096, U, 0, 0);

  // 3) gcn = relu(sum_k cheb_k @ tmp_k)  (async-LDS staged B panel)
  cheb_kernel<<<dim3(4, 5, 48), 128, 0, stream>>>(cheb, tmpx, gcnx);
  cheb_kernel<<<dim3(4, 5, 48), 128, 0, stream>>>(cheb, tmpo, gcno);

  // 4) input projections for LSTM
  launch_gemm(stream, gcnx, Ux, bx, gxb, RRT, 256, 64, 64, 256, 256, 1, 0, 0, 0, 0, 0);
  launch_gemm(stream, gcno, Ug, bg, ggb, RRT, 128, 64, 64, 128, 128, 1, 0, 0, 0, 0, 0);

  // 5) LSTM scan
  zero_kernel<<<dim3((unsigned)((2 * L64 + T256 - 1) / T256)), T256, 0, stream>>>(hb, 2 * L64);
  for (int t = 0; t < SSS; ++t)
    lstm_step_kernel<<<dim3((LLT + 15) / 16), 128, 0, stream>>>(gxb, ggb, Vx, Vg, hb, cb,
                                                                hsb, t);

  // 6) output FF
  launch_gemm(stream, hsb, Wff1, bff1, ff1b, RRT, 64, 64, 64, 64, 64, 1, 0, 0, 0, 0, 1);
  launch_gemm(stream, ff1b, Wff2, bff2, hid2, RRT, 64, 64, 64, 64, 64, 1, 0, 0, 0, 0, 0);

  // 7) time conv
  tconv_kernel<<<dim3((unsigned)((tot + T256 - 1) / T256)), T256, 0, stream>>>(hid2, Wtc,
                                                                               btc, tcb);

  // 8) attention inputs
  pack_q_kernel<<<dim3((unsigned)((L64 + T256 - 1) / T256)), T256, 0, stream>>>(tcb, qsrc);
  pack_kv_kernel<<<dim3((unsigned)((tot + T256 - 1) / T256)), T256, 0, stream>>>(ex, kvb);
  launch_gemm(stream, qsrc, Win, bin, Qhb, LLT, 64, 64, 64, 64, 64, 1, 0, 0, 0, 1, 0);
  launch_gemm(stream, kvb, Win + 64 * 64, bin + 64, Khb, LLT, 64, 64, 64, 64, 64,
              SSS, L64, 0, L64, 1, 0);
  launch_gemm(stream, kvb, Win + 128 * 64, bin + 128, Vhb, LLT, 64, 64, 64, 64, 64,
              SSS, L64, 0, L64, 1, 0);

  // 9) fused flash attention (48 instances), then output projection
  attn_kernel<<<dim3(((LLT + 15) / 16 + 3) / 4, 48), 128, 0, stream>>>(Qhb, Khb, Vhb, ctxb);
  launch_gemm(stream, ctxb, Wout, bout, dsab, RRT, 64, 64, 64, 64, 64, 1, 0, 0, 0, 1, 0);

  // 10) attention FF + residual 1x1
  launch_gemm(stream, dsab, Wfa1, bfa1, fa1b, RRT, 64, 64, 64, 64, 64, 1, 0, 0, 0, 0, 1);
  launch_gemm(stream, fa1b, Wfa2, bfa2, ffab, RRT, 64, 64, 64, 64, 64, 1, 0, 0, 0, 0, 0);
  launch_gemm(stream, hid2, Wrc, brc, resb, RRT, 64, 64, 64, 64, 64, 1, 0, 0, 0, 1, 0);

  // 11) final projection to (B,P,N)
  final_kernel<<<dim3((unsigned)(((long)BB * PPP * NNN + T256 - 1) / T256)), T256, 0,
                stream>>>(resb, ffab, Wfin, bfin, (float*)d_out);
}